// CNN2__57801669869865
// MI455X (gfx1250) — compile-verified
//
#include <hip/hip_runtime.h>
#include <cstdint>

// Reference: x (B=8, C=3, H=2048, W=2048) f32; i device scalar; (o,p)=divmod(i,2)
//            out[b,c,h,w] = x[b,c,2h+o,2w+p]  -> out (8,3,1024,1024) f32
#define TPB    256
#define W_IN   2048
#define W_OUT  1024
#define N_ROWS (8 * 3 * 1024)    // total output rows
#define RPB    4                 // output rows per block
#define NBLK   (N_ROWS / RPB)    // 6144 blocks

typedef float v4f __attribute__((ext_vector_type(4)));

__global__ __launch_bounds__(TPB) void CNN2_strided_select_kernel(
    const float* __restrict__ x,
    const int*   __restrict__ ip,
    float*       __restrict__ out)
{
    __shared__ float lds[RPB * W_IN];     // 32 KB: four full input rows

    const int t  = threadIdx.x;           // 0..255
    const int r0 = blockIdx.x * RPB;      // first output row of this block
    const int h0 = r0 & (W_OUT - 1);      // row within plane (block never crosses plane)
    const int cc = r0 >> 10;              // fused (b*C + c) plane index, 0..23

    const int i = *ip;                    // uniform scalar load
    const int o = (i >> 1) & 1;
    const int p = i & 1;

    // First input row for this block: plane cc, row 2*h0 + o.
    const size_t inRowElem = ((size_t)cc * W_IN + (size_t)(2 * h0 + o)) * (size_t)W_IN;
    const uint64_t gbase = (uint64_t)(uintptr_t)(x + inRowElem) + (uint64_t)(t * 16);
    const uint32_t lbase = (uint32_t)(uintptr_t)lds + (uint32_t)(t * 16);

    // --- Stage 4 input rows (32 KB) into LDS via CDNA5 async DMA loads ----
    // Output row r0+j reads input row (2h0+o)+2j: global stride 2*W_IN*4 = 16384 B.
    // The 24-bit inst offset applies to BOTH the LDS and global address, so
    // offset:4096 covers the second half of each 8 KB row.
    #pragma unroll
    for (int j = 0; j < RPB; ++j) {
        const uint32_t l = lbase + (uint32_t)(j * 8192);
        const uint64_t g = gbase + (uint64_t)(j * 16384);
        asm volatile("global_load_async_to_lds_b128 %0, %1, off"
                     :: "v"(l), "v"(g) : "memory");
        asm volatile("global_load_async_to_lds_b128 %0, %1, off offset:4096"
                     :: "v"(l), "v"(g) : "memory");
    }

    asm volatile("s_wait_asynccnt 0" ::: "memory");   // my 8 DMA loads done
    __syncthreads();                                   // everyone's DMA visible

    // --- Compact: per row pick elements 8t+p+{0,2,4,6} -> one float4 out ---
    #pragma unroll
    for (int j = 0; j < RPB; ++j) {
        const v4f a = *reinterpret_cast<const v4f*>(&lds[j * W_IN + 8 * t]);
        const v4f b = *reinterpret_cast<const v4f*>(&lds[j * W_IN + 8 * t + 4]);
        v4f v;
        if (p == 0) { v4f tmp = {a.x, a.z, b.x, b.z}; v = tmp; }   // uniform branch
        else        { v4f tmp = {a.y, a.w, b.y, b.w}; v = tmp; }

        v4f* dst = reinterpret_cast<v4f*>(out + (size_t)(r0 + j) * W_OUT) + t;
        __builtin_nontemporal_store(v, dst);          // stream-once output, NT hint
    }
}

extern "C" void kernel_launch(void* const* d_in, const int* in_sizes, int n_in,
                              void* d_out, int out_size, void* d_ws, size_t ws_size,
                              hipStream_t stream) {
    (void)in_sizes; (void)n_in; (void)out_size; (void)d_ws; (void)ws_size;
    const float* x   = (const float*)d_in[0];
    const int*   ip  = (const int*)d_in[1];
    float*       out = (float*)d_out;

    dim3 grid(NBLK);    // 6144 blocks, 4 output rows (16 KB out, 32 KB in) each
    dim3 block(TPB);
    CNN2_strided_select_kernel<<<grid, block, 0, stream>>>(x, ip, out);
}